// GINConv_21930103014152
// MI455X (gfx1250) — compile-verified
//
#include <hip/hip_runtime.h>

#define D 128
#define BN_EPS 1e-5f
#define LDS_STRIDE 132   // 128 + 4 pad: conflict-free ds_load_b64 (see analysis)

typedef __attribute__((ext_vector_type(2))) float v2f;
typedef __attribute__((ext_vector_type(8))) float v8f;

// ---------------------------------------------------------------------------
// Kernel 1: agg[i] = (1+eps)*x[i]  (scatter-sum accumulates on top of this)
//           + zero the BN stats accumulators.
// ---------------------------------------------------------------------------
__global__ void init_agg(const float* __restrict__ x, const float* __restrict__ eps,
                         float* __restrict__ agg,
                         float* __restrict__ colsum, float* __restrict__ colsq,
                         int total) {
  const float s = 1.0f + eps[0];
  for (int i = blockIdx.x * blockDim.x + threadIdx.x; i < total;
       i += gridDim.x * blockDim.x)
    agg[i] = s * x[i];
  if (blockIdx.x == 0 && threadIdx.x < D) {
    colsum[threadIdx.x] = 0.0f;
    colsq[threadIdx.x]  = 0.0f;
  }
}

// ---------------------------------------------------------------------------
// Kernel 2: one wave per edge: gather x[src] (32 lanes x float4 = one fully
// coalesced 512B row), ReLU, hardware non-returning f32 atomic-add into
// agg[dst]. x (20MB) and agg (20MB) stay resident in the 192MB L2, so this
// phase runs at L2 atomic/bandwidth rates, not HBM.
// ---------------------------------------------------------------------------
__global__ void edge_scatter(const float* __restrict__ x,
                             const int* __restrict__ src,
                             const int* __restrict__ dst,
                             float* __restrict__ agg, int E) {
  const int lane   = threadIdx.x & 31;
  const int wave   = (blockIdx.x * blockDim.x + threadIdx.x) >> 5;
  const int nwaves = (gridDim.x * blockDim.x) >> 5;
  for (int e = wave; e < E; e += nwaves) {
    const int s = src[e];
    const int d = dst[e];
    const float4 v = reinterpret_cast<const float4*>(x + (size_t)s * D)[lane];
    float* a = agg + (size_t)d * D + lane * 4;
    unsafeAtomicAdd(a + 0, fmaxf(v.x, 0.0f));
    unsafeAtomicAdd(a + 1, fmaxf(v.y, 0.0f));
    unsafeAtomicAdd(a + 2, fmaxf(v.z, 0.0f));
    unsafeAtomicAdd(a + 3, fmaxf(v.w, 0.0f));
  }
}

// ---------------------------------------------------------------------------
// Shared WMMA tile compute: one wave owns 4 M-subtiles (64 rows) x 16 cols.
// A read from LDS (padded stride), B (weight row, contiguous in k) from
// global (64KB weight matrix stays hot in WGP$). 4 independent accumulators
// per k-step -> no WMMA->WMMA RAW hazard, full matrix-pipe occupancy.
// ---------------------------------------------------------------------------
__device__ __forceinline__ void wmma_tile_4m(const float* __restrict__ sA,
                                             const float* __restrict__ Brow,
                                             v8f c[4], int l15, int hi) {
#pragma unroll
  for (int k = 0; k < D; k += 4) {
    const int kb = k + 2 * hi;                 // lanes 16-31 hold K=+2,+3
    const float2 bv = *reinterpret_cast<const float2*>(Brow + kb);
    v2f b; b[0] = bv.x; b[1] = bv.y;
#pragma unroll
    for (int s = 0; s < 4; ++s) {
      const float* ap = sA + (s * 16 + l15) * LDS_STRIDE + kb;
      v2f a; a[0] = ap[0]; a[1] = ap[1];       // ds_load_b64, conflict-free
      c[s] = __builtin_amdgcn_wmma_f32_16x16x4_f32(false, a, false, b,
                                                   (short)0, c[s], false, false);
    }
  }
}

// ---------------------------------------------------------------------------
// Kernel 3: h1 = agg @ W1^T + b1. Block = 8 waves = 64 rows x 128 cols.
// A-tile (64x128) staged once in LDS; fused BN batch-stat accumulation.
// ---------------------------------------------------------------------------
__global__ void gemm1_bn_stats(const float* __restrict__ A,
                               const float* __restrict__ W1,
                               const float* __restrict__ b1,
                               float* __restrict__ h1,
                               float* __restrict__ colsum,
                               float* __restrict__ colsq) {
  __shared__ float sA[64 * LDS_STRIDE];
  __shared__ float s_sum[D];
  __shared__ float s_sq[D];
  const int tileM = blockIdx.x * 64;
  const int lane  = threadIdx.x & 31;
  const int wv    = threadIdx.x >> 5;          // 0..7 : N-tile (16 cols)
  const int l15   = lane & 15;
  const int hi    = lane >> 4;

  if (threadIdx.x < D) { s_sum[threadIdx.x] = 0.0f; s_sq[threadIdx.x] = 0.0f; }
  // Cooperative A-tile fill: 64 rows x 32 float4 = 2048 float4 over 256 thr.
  for (int idx = threadIdx.x; idx < 64 * 32; idx += 256) {
    const int r  = idx >> 5;
    const int c4 = idx & 31;
    const float4 v =
        *reinterpret_cast<const float4*>(A + (size_t)(tileM + r) * D + c4 * 4);
    float* dstp = &sA[r * LDS_STRIDE + c4 * 4];
    dstp[0] = v.x; dstp[1] = v.y; dstp[2] = v.z; dstp[3] = v.w;
  }
  __syncthreads();

  const int col = wv * 16 + l15;
  const float bias = b1[col];
  v8f c[4];
#pragma unroll
  for (int s = 0; s < 4; ++s)
#pragma unroll
    for (int v = 0; v < 8; ++v) c[s][v] = bias;

  wmma_tile_4m(sA, W1 + (size_t)col * D, c, l15, hi);

  float ps = 0.0f, pq = 0.0f;
#pragma unroll
  for (int s = 0; s < 4; ++s)
#pragma unroll
    for (int v = 0; v < 8; ++v) {
      const int row = tileM + s * 16 + v + 8 * hi;  // C/D layout: VGPR v -> row v/v+8
      h1[(size_t)row * D + col] = c[s][v];
      ps += c[s][v];
      pq += c[s][v] * c[s][v];
    }
  atomicAdd(&s_sum[col], ps);
  atomicAdd(&s_sq[col], pq);
  __syncthreads();
  if (threadIdx.x < D) {
    atomicAdd(&colsum[threadIdx.x], s_sum[threadIdx.x]);
    atomicAdd(&colsq[threadIdx.x],  s_sq[threadIdx.x]);
  }
}

// ---------------------------------------------------------------------------
// Kernel 4: fold BN stats into per-column scale/shift.
// ---------------------------------------------------------------------------
__global__ void bn_finalize(const float* __restrict__ colsum,
                            const float* __restrict__ colsq,
                            const float* __restrict__ gamma,
                            const float* __restrict__ beta,
                            float* __restrict__ scale,
                            float* __restrict__ shift, int N) {
  const int t = threadIdx.x;
  if (t < D) {
    const float invN = 1.0f / (float)N;
    const float mu  = colsum[t] * invN;
    const float var = colsq[t] * invN - mu * mu;
    const float inv = rsqrtf(var + BN_EPS);
    const float sc  = gamma[t] * inv;
    scale[t] = sc;
    shift[t] = beta[t] - mu * sc;
  }
}

// ---------------------------------------------------------------------------
// Kernel 5: out = relu(h1*scale + shift) @ W2^T + b2. BN+ReLU applied once
// per element during the LDS fill; compute loop identical to GEMM1.
// ---------------------------------------------------------------------------
__global__ void gemm2(const float* __restrict__ h1,
                      const float* __restrict__ W2,
                      const float* __restrict__ b2,
                      const float* __restrict__ scale,
                      const float* __restrict__ shift,
                      float* __restrict__ out) {
  __shared__ float sA[64 * LDS_STRIDE];
  const int tileM = blockIdx.x * 64;
  const int lane  = threadIdx.x & 31;
  const int wv    = threadIdx.x >> 5;
  const int l15   = lane & 15;
  const int hi    = lane >> 4;

  for (int idx = threadIdx.x; idx < 64 * 32; idx += 256) {
    const int r  = idx >> 5;
    const int c4 = idx & 31;
    const int k0 = c4 * 4;
    float4 v =
        *reinterpret_cast<const float4*>(h1 + (size_t)(tileM + r) * D + k0);
    const float4 sc4 = *reinterpret_cast<const float4*>(scale + k0);
    const float4 sh4 = *reinterpret_cast<const float4*>(shift + k0);
    v.x = fmaxf(fmaf(v.x, sc4.x, sh4.x), 0.0f);
    v.y = fmaxf(fmaf(v.y, sc4.y, sh4.y), 0.0f);
    v.z = fmaxf(fmaf(v.z, sc4.z, sh4.z), 0.0f);
    v.w = fmaxf(fmaf(v.w, sc4.w, sh4.w), 0.0f);
    float* dstp = &sA[r * LDS_STRIDE + k0];
    dstp[0] = v.x; dstp[1] = v.y; dstp[2] = v.z; dstp[3] = v.w;
  }
  __syncthreads();

  const int col = wv * 16 + l15;
  const float bias = b2[col];
  v8f c[4];
#pragma unroll
  for (int s = 0; s < 4; ++s)
#pragma unroll
    for (int v = 0; v < 8; ++v) c[s][v] = bias;

  wmma_tile_4m(sA, W2 + (size_t)col * D, c, l15, hi);

#pragma unroll
  for (int s = 0; s < 4; ++s)
#pragma unroll
    for (int v = 0; v < 8; ++v) {
      const int row = tileM + s * 16 + v + 8 * hi;
      out[(size_t)row * D + col] = c[s][v];
    }
}

// ---------------------------------------------------------------------------
extern "C" void kernel_launch(void* const* d_in, const int* in_sizes, int n_in,
                              void* d_out, int out_size, void* d_ws, size_t ws_size,
                              hipStream_t stream) {
  const float* x     = (const float*)d_in[0];
  const int*   src   = (const int*)d_in[1];
  const int*   dst   = (const int*)d_in[2];
  const float* W1    = (const float*)d_in[3];
  const float* b1    = (const float*)d_in[4];
  const float* gamma = (const float*)d_in[5];
  const float* beta  = (const float*)d_in[6];
  const float* W2    = (const float*)d_in[7];
  const float* b2    = (const float*)d_in[8];
  const float* eps   = (const float*)d_in[9];

  const int N = in_sizes[0] / D;   // 40000
  const int E = in_sizes[1];       // 640000

  float* ws     = (float*)d_ws;
  float* agg    = ws;                               // N*D
  float* h1     = agg + (size_t)N * D;              // N*D
  float* colsum = h1  + (size_t)N * D;              // D
  float* colsq  = colsum + D;                       // D
  float* scale  = colsq  + D;                       // D
  float* shift  = scale  + D;                       // D

  const int total = N * D;
  init_agg<<<1024, 256, 0, stream>>>(x, eps, agg, colsum, colsq, total);
  edge_scatter<<<4096, 256, 0, stream>>>(x, src, dst, agg, E);
  gemm1_bn_stats<<<N / 64, 256, 0, stream>>>(agg, W1, b1, h1, colsum, colsq);
  bn_finalize<<<1, 128, 0, stream>>>(colsum, colsq, gamma, beta, scale, shift, N);
  gemm2<<<N / 64, 256, 0, stream>>>(h1, W2, b2, scale, shift, (float*)d_out);
}